// GINConv_14053132992699
// MI455X (gfx1250) — compile-verified
//
#include <hip/hip_runtime.h>
#include <hip/hip_bf16.h>

typedef __attribute__((ext_vector_type(2))) float v2f;
typedef __attribute__((ext_vector_type(8))) float v8f;

#define D_FEAT 64
#define H_DIM  256
#define TILE_M 64         // 4 sub-tiles of 16 rows -> amortize B-fragment loads 4x
#define RST_STRIDE 68     // 64 + 4 pad: A-frag reads land on distinct LDS banks
#define H_STRIDE   260    // 256 + 4 pad

// ---------------------------------------------------------------------------
// K1: rst = (1+EPS)*feat, EPS=0 -> plain copy, float4 vectorized
// ---------------------------------------------------------------------------
__global__ void gin_init_kernel(const float* __restrict__ feat,
                                float* __restrict__ rst, int n_elems4) {
    int i = blockIdx.x * blockDim.x + threadIdx.x;
    if (i < n_elems4) {
        reinterpret_cast<float4*>(rst)[i] =
            reinterpret_cast<const float4*>(feat)[i];
    }
}

// ---------------------------------------------------------------------------
// K2: scatter-add: rst[dst[e]] += feat[src[e]]  (row of 64 floats per edge)
// 16 threads per edge, each handles 4 floats -> global_atomic_add_f32
// ---------------------------------------------------------------------------
__global__ void gin_scatter_kernel(const float* __restrict__ feat,
                                   const int* __restrict__ src,
                                   const int* __restrict__ dst,
                                   float* __restrict__ rst, int n_edges) {
    long long tid = (long long)blockIdx.x * blockDim.x + threadIdx.x;
    int e = (int)(tid >> 4);
    int c = (int)(tid & 15);
    if (e >= n_edges) return;
    size_t soff = (size_t)src[e] * D_FEAT + (size_t)c * 4;
    size_t doff = (size_t)dst[e] * D_FEAT + (size_t)c * 4;
    float4 v = *reinterpret_cast<const float4*>(feat + soff);
    float* p = rst + doff;
    unsafeAtomicAdd(p + 0, v.x);   // global_atomic_add_f32 (no return)
    unsafeAtomicAdd(p + 1, v.y);
    unsafeAtomicAdd(p + 2, v.z);
    unsafeAtomicAdd(p + 3, v.w);
}

// ---------------------------------------------------------------------------
// K3: fused MLP per 64-node tile using V_WMMA_F32_16X16X4_F32
//   h   = relu(rst[64x64] @ W1[64x256] + b1)
//         16 waves, each owns one 16-wide N-slice of h, computes 4 M-tiles
//         per k-step with a single shared B fragment (4x B reuse).
//   out = h[64x256] @ W2[256x64] + b2
//         16 waves = 4 N-tiles x 4 K-splits, 4 M-tiles each; K-partials
//         reduced with ds_add_f32, epilogue adds b2.
// ---------------------------------------------------------------------------
__global__ __launch_bounds__(512)
void gin_mlp_kernel(const float* __restrict__ rst,
                    const float* __restrict__ W1,
                    const float* __restrict__ b1,
                    const float* __restrict__ W2,
                    const float* __restrict__ b2,
                    float* __restrict__ out, int n_nodes) {
    __shared__ float s_rst[TILE_M * RST_STRIDE];   // 17.4 KB
    __shared__ float s_h[TILE_M * H_STRIDE];       // 66.6 KB
    __shared__ float s_out[TILE_M * D_FEAT];       // 16   KB

    const int tid   = threadIdx.x;
    const int wave  = tid >> 5;      // 0..15 (wave32)
    const int lane  = tid & 31;
    const int node0 = blockIdx.x * TILE_M;

    // ---- stage rst tile (clamp rows so EXEC stays full for WMMA later) ----
    for (int i = tid; i < TILE_M * D_FEAT; i += 512) {
        int m = i >> 6, k = i & 63;
        int row = node0 + m;
        row = row < n_nodes ? row : (n_nodes - 1);
        s_rst[m * RST_STRIDE + k] = rst[(size_t)row * D_FEAT + k];
    }
    for (int i = tid; i < TILE_M * D_FEAT; i += 512) s_out[i] = 0.0f;
    __syncthreads();

    // fragment index helpers (ISA 7.12.2 layouts, wave32)
    const int frow  = lane & 15;           // A row / B-C-D column within tile
    const int khalf = (lane >> 4) << 1;    // lanes 0-15 -> K{0,1}; 16-31 -> K{2,3}
    const int mbase = (lane >> 4) << 3;    // C/D: VGPR r -> M=r or M=r+8

    // ------------- GEMM1: wave owns h columns [16w,16w+16), 4 M-tiles ------
    {
        const int ncol = wave * 16 + frow;
        v8f acc0 = {}, acc1 = {}, acc2 = {}, acc3 = {};
        #pragma unroll
        for (int kk = 0; kk < 16; ++kk) {
            const int k0 = kk * 4 + khalf;
            v2f b;                                    // one B frag, 4x reuse
            b.x = W1[(size_t)(k0    ) * H_DIM + ncol];
            b.y = W1[(size_t)(k0 + 1) * H_DIM + ncol];
            v2f a0, a1, a2, a3;
            a0.x = s_rst[(frow     ) * RST_STRIDE + k0];
            a0.y = s_rst[(frow     ) * RST_STRIDE + k0 + 1];
            a1.x = s_rst[(frow + 16) * RST_STRIDE + k0];
            a1.y = s_rst[(frow + 16) * RST_STRIDE + k0 + 1];
            a2.x = s_rst[(frow + 32) * RST_STRIDE + k0];
            a2.y = s_rst[(frow + 32) * RST_STRIDE + k0 + 1];
            a3.x = s_rst[(frow + 48) * RST_STRIDE + k0];
            a3.y = s_rst[(frow + 48) * RST_STRIDE + k0 + 1];
            acc0 = __builtin_amdgcn_wmma_f32_16x16x4_f32(
                false, a0, false, b, (short)0, acc0, false, false);
            acc1 = __builtin_amdgcn_wmma_f32_16x16x4_f32(
                false, a1, false, b, (short)0, acc1, false, false);
            acc2 = __builtin_amdgcn_wmma_f32_16x16x4_f32(
                false, a2, false, b, (short)0, acc2, false, false);
            acc3 = __builtin_amdgcn_wmma_f32_16x16x4_f32(
                false, a3, false, b, (short)0, acc3, false, false);
        }
        const float bias = b1[ncol];
        #pragma unroll
        for (int r = 0; r < 8; ++r) {
            float v0 = acc0[r] + bias, v1 = acc1[r] + bias;
            float v2 = acc2[r] + bias, v3 = acc3[r] + bias;
            s_h[(     mbase + r) * H_STRIDE + ncol] = v0 > 0.f ? v0 : 0.f;
            s_h[(16 + mbase + r) * H_STRIDE + ncol] = v1 > 0.f ? v1 : 0.f;
            s_h[(32 + mbase + r) * H_STRIDE + ncol] = v2 > 0.f ? v2 : 0.f;
            s_h[(48 + mbase + r) * H_STRIDE + ncol] = v3 > 0.f ? v3 : 0.f;
        }
    }
    __syncthreads();

    // ------------- GEMM2: 4 N-tiles x 4 K-splits, 4 M-tiles each -----------
    {
        const int ntile = (wave & 3) * 16;
        const int kbase = (wave >> 2) * 64;
        const int ncol  = ntile + frow;
        v8f acc0 = {}, acc1 = {}, acc2 = {}, acc3 = {};
        #pragma unroll
        for (int kk = 0; kk < 16; ++kk) {
            const int k0 = kbase + kk * 4 + khalf;
            v2f b;
            b.x = W2[(size_t)(k0    ) * D_FEAT + ncol];
            b.y = W2[(size_t)(k0 + 1) * D_FEAT + ncol];
            v2f a0, a1, a2, a3;
            a0.x = s_h[(frow     ) * H_STRIDE + k0];
            a0.y = s_h[(frow     ) * H_STRIDE + k0 + 1];
            a1.x = s_h[(frow + 16) * H_STRIDE + k0];
            a1.y = s_h[(frow + 16) * H_STRIDE + k0 + 1];
            a2.x = s_h[(frow + 32) * H_STRIDE + k0];
            a2.y = s_h[(frow + 32) * H_STRIDE + k0 + 1];
            a3.x = s_h[(frow + 48) * H_STRIDE + k0];
            a3.y = s_h[(frow + 48) * H_STRIDE + k0 + 1];
            acc0 = __builtin_amdgcn_wmma_f32_16x16x4_f32(
                false, a0, false, b, (short)0, acc0, false, false);
            acc1 = __builtin_amdgcn_wmma_f32_16x16x4_f32(
                false, a1, false, b, (short)0, acc1, false, false);
            acc2 = __builtin_amdgcn_wmma_f32_16x16x4_f32(
                false, a2, false, b, (short)0, acc2, false, false);
            acc3 = __builtin_amdgcn_wmma_f32_16x16x4_f32(
                false, a3, false, b, (short)0, acc3, false, false);
        }
        #pragma unroll
        for (int r = 0; r < 8; ++r) {          // ds_add_f32 K-reduction
            atomicAdd(&s_out[(     mbase + r) * D_FEAT + ncol], acc0[r]);
            atomicAdd(&s_out[(16 + mbase + r) * D_FEAT + ncol], acc1[r]);
            atomicAdd(&s_out[(32 + mbase + r) * D_FEAT + ncol], acc2[r]);
            atomicAdd(&s_out[(48 + mbase + r) * D_FEAT + ncol], acc3[r]);
        }
    }
    __syncthreads();

    // ---- epilogue: + b2, guarded store ----
    for (int i = tid; i < TILE_M * D_FEAT; i += 512) {
        int m = i >> 6, n = i & 63;
        int row = node0 + m;
        if (row < n_nodes)
            out[(size_t)row * D_FEAT + n] = s_out[i] + b2[n];
    }
}

// ---------------------------------------------------------------------------
extern "C" void kernel_launch(void* const* d_in, const int* in_sizes, int n_in,
                              void* d_out, int out_size, void* d_ws, size_t ws_size,
                              hipStream_t stream) {
    const float* feat = (const float*)d_in[0];
    const float* W1   = (const float*)d_in[1];
    const float* b1   = (const float*)d_in[2];
    const float* W2   = (const float*)d_in[3];
    const float* b2   = (const float*)d_in[4];
    const int*   src  = (const int*)d_in[5];
    const int*   dst  = (const int*)d_in[6];
    float*       out  = (float*)d_out;

    const int n_nodes = in_sizes[0] / D_FEAT;   // 100000
    const int n_edges = in_sizes[5];            // 1600000

    float* rst = (float*)d_ws;                  // N*64 f32 = 25.6 MB scratch

    // K1: rst = feat
    {
        int n4 = (n_nodes * D_FEAT) / 4;
        gin_init_kernel<<<(n4 + 255) / 256, 256, 0, stream>>>(feat, rst, n4);
    }
    // K2: scatter-add neighbors
    {
        long long work = (long long)n_edges * 16;
        int blocks = (int)((work + 255) / 256);
        gin_scatter_kernel<<<blocks, 256, 0, stream>>>(feat, src, dst, rst, n_edges);
    }
    // K3: fused MLP
    {
        int blocks = (n_nodes + TILE_M - 1) / TILE_M;  // 1563
        gin_mlp_kernel<<<blocks, 512, 0, stream>>>(rst, W1, b1, W2, b2, out, n_nodes);
    }
}